// MDCA_326417515027
// MI455X (gfx1250) — compile-verified
//
#include <hip/hip_runtime.h>
#include <hip/hip_bf16.h>

typedef __attribute__((ext_vector_type(2))) float v2f;
typedef __attribute__((ext_vector_type(8))) float v8f;

#define BATCH   8192
#define CLASSES 10000
#define LOG2E   1.4426950408889634f
#define ROWS_PER_CHUNK 512
#define WAVES_PER_BLOCK 8
#define N_TILES ((CLASSES + 15) / 16)   // 625

// ---------------- Pass 1: per-row max and 1/(B*sumexp) ----------------
__global__ __launch_bounds__(256) void row_stats_kernel(
    const float* __restrict__ x, float* __restrict__ rowmax,
    float* __restrict__ rowscale) {
  __shared__ float rowc[CLASSES];
  __shared__ float red[8];
  const int row = blockIdx.x;
  const int tid = threadIdx.x;
  const float* rp = x + (size_t)row * CLASSES;

  float m = -3.402823466e38f;
  for (int c = tid; c < CLASSES; c += 256) {
    float v = rp[c];
    rowc[c] = v;
    m = fmaxf(m, v);
  }
  #pragma unroll
  for (int off = 16; off >= 1; off >>= 1) m = fmaxf(m, __shfl_xor(m, off, 32));
  if ((tid & 31) == 0) red[tid >> 5] = m;
  __syncthreads();
  float M = red[0];
  #pragma unroll
  for (int i = 1; i < 8; ++i) M = fmaxf(M, red[i]);
  __syncthreads();

  float s = 0.f;
  for (int c = tid; c < CLASSES; c += 256)
    s += __builtin_amdgcn_exp2f((rowc[c] - M) * LOG2E);
  #pragma unroll
  for (int off = 16; off >= 1; off >>= 1) s += __shfl_xor(s, off, 32);
  if ((tid & 31) == 0) red[tid >> 5] = s;
  __syncthreads();
  if (tid == 0) {
    float S = 0.f;
    #pragma unroll
    for (int i = 0; i < 8; ++i) S += red[i];
    rowmax[row]   = M;
    rowscale[row] = 1.0f / ((float)BATCH * S);
  }
}

// ------- Pass 2: scaled column reduction via V_WMMA_F32_16X16X4_F32 -------
// D[m,n] = sum_k A[m,k]*B[k,n];  A[m,k] = s_{r0+k} (broadcast over m),
// B[k,n] = exp(x[r0+k, c0+n] - m_{r0+k})  =>  D row = partial avg_conf.
__global__ __launch_bounds__(256) void colsum_wmma_kernel(
    const float* __restrict__ x, const float* __restrict__ rowmax,
    const float* __restrict__ rowscale, float* __restrict__ avg_conf) {
  __shared__ float smax[ROWS_PER_CHUNK];
  __shared__ float sscale[ROWS_PER_CHUNK];
  const int tid = threadIdx.x;
  const int r0  = blockIdx.y * ROWS_PER_CHUNK;
  for (int i = tid; i < ROWS_PER_CHUNK; i += 256) {
    smax[i]   = rowmax[r0 + i];
    sscale[i] = rowscale[r0 + i];
  }
  __syncthreads();

  const int wave = tid >> 5;
  const int lane = tid & 31;
  const int tile = blockIdx.x * WAVES_PER_BLOCK + wave;
  if (tile >= N_TILES) return;            // wave-uniform; EXEC stays full for WMMA
  const int c0   = tile * 16;
  const int half = lane >> 4;             // K half: lanes 0-15 -> K=0,1 ; 16-31 -> K=2,3
  const int ln   = lane & 15;             // column within tile

  v8f acc = {};
  const float* base = x + (size_t)r0 * CLASSES + (size_t)(c0 + ln);
  for (int k = 0; k < ROWS_PER_CHUNK; k += 4) {
    const int kb = k + 2 * half;
    const float x0 = base[(size_t)kb * CLASSES];
    const float x1 = base[(size_t)(kb + 1) * CLASSES];
    v2f b, a;
    b.x = __builtin_amdgcn_exp2f((x0 - smax[kb])     * LOG2E);
    b.y = __builtin_amdgcn_exp2f((x1 - smax[kb + 1]) * LOG2E);
    a.x = sscale[kb];
    a.y = sscale[kb + 1];
    acc = __builtin_amdgcn_wmma_f32_16x16x4_f32(
        /*neg_a=*/false, a, /*neg_b=*/false, b,
        /*c_mod=*/(short)0, acc, /*reuse_a=*/false, /*reuse_b=*/false);
  }
  // All D rows are identical; lane L (<16) holds D[0, L] in acc[0].
  if (lane < 16) atomicAdd(&avg_conf[c0 + ln], acc[0]);
}

// ---------------- Histogram of targets (fraction accumulated later) ----------------
__global__ void hist_kernel(const int* __restrict__ t, float* __restrict__ counts) {
  int i = blockIdx.x * blockDim.x + threadIdx.x;
  if (i < BATCH) atomicAdd(&counts[t[i]], 1.0f);
}

// ---------------- Final L1 reduction ----------------
__global__ __launch_bounds__(1024) void final_kernel(
    const float* __restrict__ avg_conf, const float* __restrict__ counts,
    float* __restrict__ out) {
  __shared__ float red[32];
  const int tid = threadIdx.x;
  float s = 0.f;
  for (int c = tid; c < CLASSES; c += 1024)
    s += fabsf(avg_conf[c] - counts[c] * (1.0f / (float)BATCH));
  #pragma unroll
  for (int off = 16; off >= 1; off >>= 1) s += __shfl_xor(s, off, 32);
  if ((tid & 31) == 0) red[tid >> 5] = s;
  __syncthreads();
  if (tid < 32) {
    float v = red[tid];
    #pragma unroll
    for (int off = 16; off >= 1; off >>= 1) v += __shfl_xor(v, off, 32);
    if (tid == 0) out[0] = v * (1.0f / (float)CLASSES);
  }
}

extern "C" void kernel_launch(void* const* d_in, const int* in_sizes, int n_in,
                              void* d_out, int out_size, void* d_ws, size_t ws_size,
                              hipStream_t stream) {
  const float* x   = (const float*)d_in[0];
  const int*   tgt = (const int*)d_in[1];
  float* out = (float*)d_out;

  char* ws = (char*)d_ws;
  float* rowmax   = (float*)(ws);                     // 8192 f32
  float* rowscale = (float*)(ws + 32768);             // 8192 f32
  float* avg_conf = (float*)(ws + 65536);             // 10000 f32
  float* counts   = (float*)(ws + 65536 + 40064);     // 10000 f32

  // zero both accumulators (contiguous span); capturable as a graph memset node
  hipMemsetAsync(ws + 65536, 0, 40064 + 40000, stream);

  row_stats_kernel<<<BATCH, 256, 0, stream>>>(x, rowmax, rowscale);

  dim3 grid((N_TILES + WAVES_PER_BLOCK - 1) / WAVES_PER_BLOCK,
            BATCH / ROWS_PER_CHUNK);                  // 79 x 16
  colsum_wmma_kernel<<<grid, 256, 0, stream>>>(x, rowmax, rowscale, avg_conf);

  hist_kernel<<<BATCH / 256, 256, 0, stream>>>(tgt, counts);

  final_kernel<<<1, 1024, 0, stream>>>(avg_conf, counts, out);
}